// CubicFeatureSampling_32040456028956
// MI455X (gfx1250) — compile-verified
//
#include <hip/hip_runtime.h>
#include <cstdint>
#include <cstddef>

typedef __attribute__((ext_vector_type(4))) float f32x4;

#define C_DIM 64
#define S_DIM 32
#define V_DIM (S_DIM * S_DIM * S_DIM)   // 32768 voxels

// ---------------------------------------------------------------------------
// Kernel 1: transpose feature volume [B][C][V] -> workspace [B][V][C]
// Stages 64x64 (C x V) tiles in LDS using gfx1250 async global->LDS b128 DMA
// (ASYNCcnt path), then writes channel-last with coalesced b128 stores.
// ---------------------------------------------------------------------------
__global__ void __launch_bounds__(256)
cfs_transpose_kernel(const float* __restrict__ cf, float* __restrict__ ws) {
  __shared__ float tile[C_DIM][64 + 1];          // +1 pad: conflict-free column reads
  const int tid = threadIdx.x;                   // 256 threads = 8 waves
  const int blk = blockIdx.x;                    // B * (V_DIM/64) blocks
  const int b   = blk >> 9;                      // V_DIM/64 == 512
  const int v0  = (blk & 511) << 6;
  const float* src = cf + (size_t)b * C_DIM * V_DIM + v0;

  // Async copy: 64 channels x 64 voxels = 1024 float4 chunks, 4 per thread.
#pragma unroll
  for (int i = 0; i < 4; ++i) {
    const int li = tid + i * 256;                // 0..1023
    const int c  = li >> 4;                      // 16 float4 per channel row
    const int vv = (li & 15) << 2;
    const float*   gp = src + (size_t)c * V_DIM + vv;
    const uint32_t lp = (uint32_t)(uintptr_t)&tile[c][vv];  // low 32b = LDS byte addr
    asm volatile("global_load_async_to_lds_b128 %0, %1, off"
                 :: "v"(lp), "v"(gp) : "memory");
  }
  asm volatile("s_wait_asynccnt 0" ::: "memory");
  __syncthreads();

  // Write out transposed: ws[b][v][c], 64 contiguous channels per voxel.
  float* dst = ws + ((size_t)b * V_DIM + v0) * C_DIM;
#pragma unroll
  for (int i = 0; i < 4; ++i) {
    const int li = tid + i * 256;                // 0..1023 float4 chunks
    const int vv = li >> 4;
    const int c4 = (li & 15) << 2;
    f32x4 v;
    v.x = tile[c4 + 0][vv];
    v.y = tile[c4 + 1][vv];
    v.z = tile[c4 + 2][vv];
    v.w = tile[c4 + 3][vv];
    *(f32x4*)(dst + (size_t)vv * C_DIM + c4) = v;
  }
}

// ---------------------------------------------------------------------------
// Shared index math for the gather kernels.
// ---------------------------------------------------------------------------
__device__ __forceinline__ void cfs_indices(const float* __restrict__ pts,
                                            unsigned long long pn, unsigned corner,
                                            bool& valid, unsigned& flat) {
  const float px = pts[pn * 3 + 0];
  const float py = pts[pn * 3 + 1];
  const float pz = pts[pn * 3 + 2];
  const float h = (float)S_DIM * 0.5f;
  const int lx = (int)floorf(px * h + h);
  const int ly = (int)floorf(py * h + h);
  const int lz = (int)floorf(pz * h + h);
  const int cx = lx + (int)((corner >> 2) & 1u);
  const int cy = ly + (int)((corner >> 1) & 1u);
  const int cz = lz + (int)(corner & 1u);
  valid = ((unsigned)cx < (unsigned)S_DIM) &
          ((unsigned)cy < (unsigned)S_DIM) &
          ((unsigned)cz < (unsigned)S_DIM);
  const int ix = min(max(cx, 0), S_DIM - 1);
  const int iy = min(max(cy, 0), S_DIM - 1);
  const int iz = min(max(cz, 0), S_DIM - 1);
  flat = ((unsigned)ix * S_DIM + (unsigned)iy) * S_DIM + (unsigned)iz;
}

// ---------------------------------------------------------------------------
// Kernel 2: main gather. One thread per (point, corner, 4-channel group).
// Wave32 covers exactly one point (uniform point load); each half-wave reads
// and writes a contiguous 256B segment. Output stored non-temporally so the
// 512MB stream doesn't evict the L2-resident 32MB volume.
// nShift = log2(N) when N is a power of two (avoids runtime u64 division).
// ---------------------------------------------------------------------------
__global__ void __launch_bounds__(256)
cfs_gather_kernel(const float* __restrict__ pts, const float* __restrict__ vol,
                  float* __restrict__ out, unsigned int BN, unsigned int N,
                  int nShift) {
  const long long gid = (long long)blockIdx.x * 256 + threadIdx.x;
  const unsigned int c4     = (unsigned int)(gid & 15);
  const unsigned int corner = (unsigned int)((gid >> 4) & 7);
  const unsigned long long pn = (unsigned long long)(gid >> 7);
  if (pn >= BN) return;
  const unsigned int b = (nShift >= 0) ? (unsigned int)(pn >> nShift)
                                       : (unsigned int)(pn / N);

  bool valid; unsigned flat;
  cfs_indices(pts, pn, corner, valid, flat);

  const f32x4* src =
      (const f32x4*)(vol + ((size_t)b * V_DIM + flat) * C_DIM) + c4;
  f32x4 v = *src;
  v *= (valid ? 1.0f : 0.0f);

  f32x4* dst = (f32x4*)(out + ((size_t)pn * 8 + corner) * C_DIM) + c4;
  __builtin_nontemporal_store(v, dst);
}

// ---------------------------------------------------------------------------
// Fallback: gather directly from channel-major layout (if ws too small).
// ---------------------------------------------------------------------------
__global__ void __launch_bounds__(256)
cfs_gather_direct_kernel(const float* __restrict__ pts, const float* __restrict__ cf,
                         float* __restrict__ out, unsigned int BN, unsigned int N,
                         int nShift) {
  const long long gid = (long long)blockIdx.x * 256 + threadIdx.x;
  const unsigned int c4     = (unsigned int)(gid & 15);
  const unsigned int corner = (unsigned int)((gid >> 4) & 7);
  const unsigned long long pn = (unsigned long long)(gid >> 7);
  if (pn >= BN) return;
  const unsigned int b = (nShift >= 0) ? (unsigned int)(pn >> nShift)
                                       : (unsigned int)(pn / N);

  bool valid; unsigned flat;
  cfs_indices(pts, pn, corner, valid, flat);

  const float* src = cf + (size_t)b * C_DIM * V_DIM + flat;
  f32x4 v;
#pragma unroll
  for (int k = 0; k < 4; ++k)
    v[k] = src[(size_t)(c4 * 4 + k) * V_DIM];
  v *= (valid ? 1.0f : 0.0f);

  f32x4* dst = (f32x4*)(out + ((size_t)pn * 8 + corner) * C_DIM) + c4;
  __builtin_nontemporal_store(v, dst);
}

// ---------------------------------------------------------------------------
extern "C" void kernel_launch(void* const* d_in, const int* in_sizes, int n_in,
                              void* d_out, int out_size, void* d_ws, size_t ws_size,
                              hipStream_t stream) {
  // d_in[0] = scale (unused; geometry fixed at S=32, C=64 per reference)
  const float* pts = (const float*)d_in[1];
  const float* cf  = (const float*)d_in[2];
  float* out = (float*)d_out;

  const unsigned int BN = (unsigned int)(in_sizes[1] / 3);                  // B*N
  const unsigned int B  = (unsigned int)(in_sizes[2] / (C_DIM * V_DIM));
  const unsigned int N  = BN / B;

  // log2(N) if N is a power of two, else -1 (kernel falls back to division)
  int nShift = -1;
  if (N && !(N & (N - 1))) {
    nShift = 0;
    while ((1u << nShift) != N) ++nShift;
  }

  const long long total = (long long)BN * 128;   // 8 corners * 16 float4 groups
  const unsigned int gblocks = (unsigned int)((total + 255) / 256);

  const size_t need = (size_t)in_sizes[2] * sizeof(float);
  if (ws_size >= need) {
    cfs_transpose_kernel<<<B * (V_DIM / 64), 256, 0, stream>>>(cf, (float*)d_ws);
    cfs_gather_kernel<<<gblocks, 256, 0, stream>>>(pts, (float*)d_ws, out, BN, N, nShift);
  } else {
    cfs_gather_direct_kernel<<<gblocks, 256, 0, stream>>>(pts, cf, out, BN, N, nShift);
  }
}